// MXQuantizer_39814346834454
// MI455X (gfx1250) — compile-verified
//
#include <hip/hip_runtime.h>
#include <stdint.h>

// MX fp8_e4m3 fake-quant for MI455X (gfx1250, wave32).
// Memory-bound streaming kernel: 512 MB traffic -> ~22 us floor @ 23.3 TB/s.
// CDNA5 paths: global_load_async_to_lds_b128 (depth-4 ASYNCcnt software
// pipeline, constant-immediate s_wait_asynccnt), ds_swizzle_b32 wave32 group
// reduction, NT stores so the 256 MB single-use stream doesn't thrash L2.

typedef float v4f __attribute__((ext_vector_type(4)));
typedef int   v4i __attribute__((ext_vector_type(4)));

#define TPB   256
#define NBLK  2048
#define DEPTH 4   // async pipeline depth (LDS: (4+1)*256*16B = 20 KB of 320 KB)

// ---------------- CDNA5 async global->LDS copy ----------------
// Builtin signature (from hipcc diagnostic): param0 is AS1 (__device__) int4*.
typedef __attribute__((address_space(1))) v4i* g4i_t;
typedef __attribute__((address_space(3))) v4i* l4i_t;

__device__ __forceinline__ void async_copy_g2l_b128(const v4f* gsrc, const v4f* lds_dst) {
#if __has_builtin(__builtin_amdgcn_global_load_async_to_lds_b128)
  // inttoptr casts: generic global addr == AS1 addr; low 32 bits of a generic
  // shared-variable address == LDS byte offset == AS3 pointer value.
  g4i_t g = (g4i_t)(uintptr_t)gsrc;
  l4i_t l = (l4i_t)(uint32_t)(uintptr_t)lds_dst;
  __builtin_amdgcn_global_load_async_to_lds_b128(g, l, /*offset=*/0, /*cpol=*/0);
#else
  uint32_t lds_off = (uint32_t)(uintptr_t)lds_dst;
  asm volatile("global_load_async_to_lds_b128 %0, %1, off"
               :: "v"(lds_off), "v"(gsrc) : "memory");
#endif
}

template <int N>
__device__ __forceinline__ void wait_async_imm() {
#if __has_builtin(__builtin_amdgcn_s_wait_asynccnt)
  __builtin_amdgcn_s_wait_asynccnt(N);
#else
  asm volatile("s_wait_asynccnt %0" :: "n"(N) : "memory");
#endif
}

// ds_swizzle_b32 XOR swap, group-of-32 mode: offset = {1'b0, xor[4:0], or=0, and=0x1F}.
// Single DS op, no lane-index VALU (vs __shfl_xor's mbcnt/cndmask sequence).
template <int IMM>
__device__ __forceinline__ float swz(float v) {
  return __int_as_float(__builtin_amdgcn_ds_swizzle(__float_as_int(v), IMM));
}

// ---------------- e4m3 fake-quant math (no transcendentals) ----------------
// Reference per element (MBITS-2 == 3):
//   pe = max(floor(log2(|v| + (v==0))), -6); sc = 2^(3-pe)
//   q  = clip(sign(v*sc)*floor(|v*sc|+0.5)/sc, +-448)
// Bit form: eb = biased exp of v; eb==0 (zero/denorm) clamps identically to pe=-6.
__device__ __forceinline__ float qdq_elem(float xv, float rscale, float scale) {
  float v  = xv * rscale;
  int   eb = (__float_as_int(v) >> 23) & 0xFF;
  eb = (eb > 121) ? eb : 121;                        // pe = max(pe, -6)
  float sc  = __int_as_float((257 - eb) << 23);      // 2^(3-pe),  pe = eb-127
  float isc = __int_as_float((eb  - 3) << 23);       // exact 1/sc (power of two)
  float o = v * sc;                                  // |o| in [8,16)
  float r = floorf(fabsf(o) + 0.5f);                 // round half away from zero
  r = copysignf(r, v);
  float q = r * isc;
  q = fminf(fmaxf(q, -448.0f), 448.0f);              // saturate e4m3 max-normal
  return q * scale;
}

// One float4 per lane; MX group of 32 = 8 consecutive lanes (wave32).
__device__ __forceinline__ v4f quantize_group(v4f v) {
  float ax = fmaxf(fmaxf(fabsf(v.x), fabsf(v.y)), fmaxf(fabsf(v.z), fabsf(v.w)));
  ax = fmaxf(ax, swz<0x041F>(ax));                   // xor 1
  ax = fmaxf(ax, swz<0x081F>(ax));                   // xor 2
  ax = fmaxf(ax, swz<0x101F>(ax));                   // xor 4
  // shared_exp = floor(log2(max)) - 8, clamped at -127 (upper clamp +128
  // unreachable: eb<=254 -> se<=119). eb==0 (max==0/denorm) -> se=-127, same
  // as reference's log2(max + 2^-126*(max==0)) path after clamping.
  int eb = (__float_as_int(ax) >> 23) & 0xFF;
  int se = eb - 135;
  se = (se > -127) ? se : -127;
  float scale  = fmaxf(ldexpf(1.0f, se), 1e-5f);     // v_ldexp_f32, denorm ok
  // scale is 2^se unless clamped (se <= -17): pick exact po2 reciprocal else 1e5.
  float rscale = (se >= -16) ? ldexpf(1.0f, -se) : 1e5f;
  v4f r;
  r.x = qdq_elem(v.x, rscale, scale);
  r.y = qdq_elem(v.y, rscale, scale);
  r.z = qdq_elem(v.z, rscale, scale);
  r.w = qdq_elem(v.w, rscale, scale);
  return r;
}

// ---------------- main kernel: depth-4 async pipeline ----------------
// Pipeline invariant: exactly DEPTH-1 issues precede the loop and one issue
// (real or dummy) happens per iteration, so before the wait at iteration `it`
// exactly (DEPTH-1)+(it+1) loads were issued in order. s_wait_asynccnt DEPTH-1
// therefore guarantees the oldest it+1 loads (all real tiles 0..it) landed.
__global__ void __launch_bounds__(TPB)
mxq_async_kernel(const float* __restrict__ x, float* __restrict__ y, int iters) {
  __shared__ v4f smem[DEPTH][TPB];   // per-thread private slots -> no barriers
  __shared__ v4f scratch[TPB];       // dummy target for drain-phase issues
  const int tid = threadIdx.x;
  const long long stride = (long long)NBLK * TPB;            // in float4 units
  long long idx  = (long long)blockIdx.x * TPB + tid;
  long long pidx = idx;
  const v4f* __restrict__ x4 = (const v4f*)x;
  v4f* __restrict__       y4 = (v4f*)y;

  for (int p = 0; p < DEPTH - 1; ++p) {                      // prologue
    if (p < iters) { async_copy_g2l_b128(x4 + pidx, &smem[p][tid]); pidx += stride; }
    else           { async_copy_g2l_b128(x4 + idx,  &scratch[tid]); }
  }
  for (int it = 0; it < iters; ++it) {
    if (it + DEPTH - 1 < iters) {                            // keep pipe full
      async_copy_g2l_b128(x4 + pidx, &smem[(it + DEPTH - 1) & (DEPTH - 1)][tid]);
      pidx += stride;
    } else {                                                 // drain: dummy issue
      async_copy_g2l_b128(x4 + idx, &scratch[tid]);
    }
    wait_async_imm<DEPTH - 1>();                             // oldest tile done
    v4f v = smem[it & (DEPTH - 1)][tid];                     // ds_load_b128
    v4f q = quantize_group(v);
    __builtin_nontemporal_store(q, y4 + idx);                // NT: don't pollute L2
    idx += stride;
  }
}

// ---------------- fallback for sizes not divisible by NBLK*TPB*4 ----------------
__global__ void __launch_bounds__(TPB)
mxq_plain_kernel(const float* __restrict__ x, float* __restrict__ y, long long n4) {
  long long idx = (long long)blockIdx.x * TPB + threadIdx.x;
  bool valid = idx < n4;
  v4f v = {0.0f, 0.0f, 0.0f, 0.0f};                          // |0| contributes nothing
  if (valid) v = __builtin_nontemporal_load((const v4f*)x + idx);
  v4f q = quantize_group(v);                                 // all lanes swizzle
  if (valid) __builtin_nontemporal_store(q, (v4f*)y + idx);
}

extern "C" void kernel_launch(void* const* d_in, const int* in_sizes, int n_in,
                              void* d_out, int out_size, void* d_ws, size_t ws_size,
                              hipStream_t stream) {
  const float* x = (const float*)d_in[0];
  float*       y = (float*)d_out;
  long long n  = in_sizes[0];      // 8192*8192 for the reference shape
  long long n4 = n >> 2;           // MX group=32 -> n is a multiple of 4
  const long long tile = (long long)NBLK * TPB;
  if (n4 > 0 && (n4 % tile) == 0) {
    mxq_async_kernel<<<NBLK, TPB, 0, stream>>>(x, y, (int)(n4 / tile));  // 32 iters @ 8192^2
  } else if (n4 > 0) {
    long long blocks = (n4 + TPB - 1) / TPB;
    mxq_plain_kernel<<<(int)blocks, TPB, 0, stream>>>(x, y, n4);
  }
}